// ScoringFunction_1675037245543
// MI455X (gfx1250) — compile-verified
//
#include <hip/hip_runtime.h>

typedef __attribute__((ext_vector_type(2))) float v2f;
typedef __attribute__((ext_vector_type(8))) float v8f;

#define NUM_NODES 100000
#define INPUT_DIM 256
#define NUM_BAGS  2048
#define BAG_SIZE  64
#define N_EDGES   3200000

// ---------------- workspace zeroing (ws is not re-poisoned between replays) ----
__global__ void zero_kernel(float* __restrict__ p, int n) {
    int i = blockIdx.x * blockDim.x + threadIdx.x;
    if (i < n) p[i] = 0.0f;
}

// ---------------- edge aggregation: nbr_sum + degree via L2 float atomics ------
__global__ void edge_kernel(const int* __restrict__ edge_src,
                            const int* __restrict__ edge_nbr,
                            const float* __restrict__ node_weights,
                            float* __restrict__ nbr_sum,
                            float* __restrict__ deg) {
    int i = blockIdx.x * blockDim.x + threadIdx.x;
    if (i < N_EDGES) {
        int s  = edge_src[i];
        int nb = edge_nbr[i];
        atomicAdd(&nbr_sum[s], node_weights[nb]);
        atomicAdd(&deg[s], 1.0f);
    }
}

// ---------------- scoring: gather rows, fp32 WMMA dot, scale, reduce -----------
// Block = 128 threads = 4 waves; wave w handles bag elements [16w, 16w+16).
// V_WMMA_F32_16X16X4_F32 layout (ISA 7.12.2):
//   A 16x4 f32: vgpr0 = K=0 (lanes 0-15) / K=2 (lanes 16-31); vgpr1 = K=1 / K=3
//   B 4x16 f32: vgpr0 = K=0 row (lower) / K=2 row (upper);    vgpr1 = K=1 / K=3
//   D 16x16 f32: vgpr j = row j (lanes 0-15) / row j+8 (lanes 16-31)
// B columns are all theta -> D[m][*] == h[m], so lane0/lane16 hold h[0..7]/h[8..15].
__global__ __launch_bounds__(128) void scoring_kernel(
        const float* __restrict__ x,
        const int*   __restrict__ bags,
        const float* __restrict__ alpha,
        const float* __restrict__ theta,
        const float* __restrict__ nbr_sum,
        const float* __restrict__ deg,
        float*       __restrict__ out) {
    __shared__ float s_theta[INPUT_DIM];
    __shared__ float s_h[BAG_SIZE];
    __shared__ float s_wave[4];

    int t = threadIdx.x;
    s_theta[t]        = theta[t];
    s_theta[t + 128]  = theta[t + 128];
    __syncthreads();

    int b    = blockIdx.x;
    int wave = t >> 5;
    int lane = t & 31;
    int m    = lane & 15;     // row within the wave's 16x16 tile
    int half = lane >> 4;     // K-pair selector (0 -> K0/K1, 1 -> K2/K3)
    int elem = wave * 16 + m;
    int node = bags[b * BAG_SIZE + elem];

    const float* row = x + (long)node * INPUT_DIM + half * 2;
    const float* th  = s_theta + half * 2;

    v8f c = {0.f, 0.f, 0.f, 0.f, 0.f, 0.f, 0.f, 0.f};
#pragma unroll 8
    for (int kk = 0; kk < INPUT_DIM; kk += 4) {
        v2f a = *(const v2f*)(row + kk);   // 8B-aligned global_load_b64
        v2f bb;
        bb.x = th[kk];
        bb.y = th[kk + 1];
        c = __builtin_amdgcn_wmma_f32_16x16x4_f32(
                /*neg_a=*/false, a, /*neg_b=*/false, bb,
                /*c_mod=*/(short)0, c, /*reuse_a=*/false, /*reuse_b=*/false);
    }

    // Extract h: lane 0 holds h[0..7] in c[0..7]; lane 16 holds h[8..15].
    if (m == 0) {
#pragma unroll
        for (int j = 0; j < 8; ++j)
            s_h[wave * 16 + half * 8 + j] = c[j];
    }
    __syncthreads();

    float partial = 0.0f;
    if (lane < 16) {   // here m == lane, so `node` matches element wave*16+lane
        float h  = s_h[wave * 16 + lane];
        float d  = deg[node];
        float ns = (d > 0.0f) ? nbr_sum[node] : 1.0f;
        float al = alpha[b * BAG_SIZE + elem];
        partial  = h * al * ns;
    }
    // wave32 butterfly reduction (lanes >= 16 contribute 0)
#pragma unroll
    for (int off = 16; off >= 1; off >>= 1)
        partial += __shfl_xor(partial, off);
    if (lane == 0) s_wave[wave] = partial;
    __syncthreads();
    if (t == 0) out[b] = s_wave[0] + s_wave[1] + s_wave[2] + s_wave[3];
}

extern "C" void kernel_launch(void* const* d_in, const int* in_sizes, int n_in,
                              void* d_out, int out_size, void* d_ws, size_t ws_size,
                              hipStream_t stream) {
    const float* x            = (const float*)d_in[0];
    const int*   bags         = (const int*)  d_in[1];
    const float* alpha        = (const float*)d_in[2];
    const int*   edge_src     = (const int*)  d_in[3];
    const int*   edge_nbr     = (const int*)  d_in[4];
    const float* node_weights = (const float*)d_in[5];
    const float* theta        = (const float*)d_in[6];

    float* nbr_sum = (float*)d_ws;
    float* deg     = nbr_sum + NUM_NODES;
    float* out     = (float*)d_out;

    zero_kernel<<<(2 * NUM_NODES + 255) / 256, 256, 0, stream>>>(nbr_sum, 2 * NUM_NODES);
    edge_kernel<<<(N_EDGES + 255) / 256, 256, 0, stream>>>(edge_src, edge_nbr,
                                                           node_weights, nbr_sum, deg);
    scoring_kernel<<<NUM_BAGS, 128, 0, stream>>>(x, bags, alpha, theta,
                                                 nbr_sum, deg, out);
}